// HashEncoder_27539330302144
// MI455X (gfx1250) — compile-verified
//
#include <hip/hip_runtime.h>
#include <cmath>

// Native 4-wide float vector (works with scalar splat ops and
// __builtin_nontemporal_store, unlike HIP's struct float4).
typedef float v4f __attribute__((ext_vector_type(4)));

typedef __attribute__((address_space(1))) int* gint_ptr;   // global (AS1)
typedef __attribute__((address_space(3))) int* lint_ptr;   // LDS    (AS3)

struct LevelMeta {
    float scale[4];
    int   res[4];
    int   off[4];    // level base offset, in float4 (table-row) units
};

// CDNA5 async global->LDS path (ASYNCcnt). Guarded so the file compiles on
// toolchains that don't expose the gfx1250 async builtins.
#if defined(__has_builtin)
#if __has_builtin(__builtin_amdgcn_global_load_async_to_lds_b32) && \
    __has_builtin(__builtin_amdgcn_s_wait_asynccnt)
#define USE_ASYNC_LDS 1
#endif
#endif

__global__ void __launch_bounds__(256)
hashgrid_encode_kernel(const float* __restrict__ pos,
                       const v4f*  __restrict__ tab,
                       v4f*        __restrict__ out,
                       LevelMeta meta, int npoints)
{
    __shared__ float spos[256 * 3];

    const int  b0   = blockIdx.x << 8;            // first point of this block
    const int  t    = threadIdx.x;
    const float* src = pos + (size_t)b0 * 3;
    const int  nload = (npoints - b0) >= 256 ? 768 : (npoints - b0) * 3;

#ifdef USE_ASYNC_LDS
    // Stage 768 dwords of positions straight into LDS via the async data path.
    for (int i = t; i < nload; i += 256) {
        __builtin_amdgcn_global_load_async_to_lds_b32(
            (gint_ptr)(src + i),
            (lint_ptr)(&spos[i]),
            /*offset=*/0, /*cpol=*/0);
    }
    __builtin_amdgcn_s_wait_asynccnt(0);   // this wave's async copies done
    __syncthreads();                       // all waves' copies visible
#else
    for (int i = t; i < nload; i += 256) spos[i] = src[i];
    __syncthreads();
#endif

    const int n = b0 + t;
    if (n >= npoints) return;

    // stride-3 LDS reads: 3t mod 64 is a bijection over lanes -> conflict-free
    const float px = spos[t * 3 + 0];
    const float py = spos[t * 3 + 1];
    const float pz = spos[t * 3 + 2];

    v4f result[4];

#pragma unroll
    for (int L = 0; L < 4; ++L) {
        const float s  = meta.scale[L];
        const int   R  = meta.res[L];
        const int   R2 = R * R;

        const float x = px * s + 0.5f;
        const float y = py * s + 0.5f;
        const float z = pz * s + 0.5f;

        const float gx = floorf(x), gy = floorf(y), gz = floorf(z);
        const float dx = x - gx,    dy = y - gy,    dz = z - gz;
        const float ox = 1.0f - dx, oy = 1.0f - dy, oz = 1.0f - dz;

        const int ix = (int)gx, iy = (int)gy, iz = (int)gz;
        const int base = ix + iy * R + iz * R2;

        // 8 corner gathers: 16B rows, 16B aligned -> global_load_b128 each.
        const v4f* __restrict__ tb = tab + meta.off[L];
        const v4f c000 = tb[base];
        const v4f c100 = tb[base + 1];
        const v4f c010 = tb[base + R];
        const v4f c110 = tb[base + R + 1];
        const v4f c001 = tb[base + R2];
        const v4f c101 = tb[base + R2 + 1];
        const v4f c011 = tb[base + R2 + R];
        const v4f c111 = tb[base + R2 + R + 1];

        // Corner order c = 0..7 (bit0=x, bit1=y, bit2=z), weight product
        // associated x->y->z, matching the reference accumulation order.
        v4f acc = (ox * oy * oz) * c000;
        acc    += (dx * oy * oz) * c100;
        acc    += (ox * dy * oz) * c010;
        acc    += (dx * dy * oz) * c110;
        acc    += (ox * oy * dz) * c001;
        acc    += (dx * oy * dz) * c101;
        acc    += (ox * dy * dz) * c011;
        acc    += (dx * dy * dz) * c111;
        result[L] = acc;
    }

    // 64B/point streaming output: non-temporal so the 128MB stream does not
    // evict the 11MB table from the 192MB L2.
    v4f* o = out + (size_t)n * 4;
#pragma unroll
    for (int L = 0; L < 4; ++L)
        __builtin_nontemporal_store(result[L], o + L);
}

static LevelMeta make_meta()
{
    // Deterministic double-precision replication of the reference's
    // _level_meta() / _encode() constants.
    const double B_SCALE    = 1.3195079565048218;
    const double BASE       = 32.0;
    const long long MAXP    = 1LL << 19;

    LevelMeta m;
    long long off = 0;
    for (int i = 0; i < 4; ++i) {
        const double scale = BASE * std::pow(B_SCALE, (double)i) - 1.0;
        const int res = (int)std::ceil(scale) + 1;
        long long p = (long long)res * res * res;
        if (p % 8) p = (p + 7) / 8 * 8;
        if (p > MAXP) p = MAXP;
        m.scale[i] = (float)scale;
        m.res[i]   = res;
        m.off[i]   = (int)off;    // in table rows (float4 units)
        off += p;
    }
    return m;
}

extern "C" void kernel_launch(void* const* d_in, const int* in_sizes, int n_in,
                              void* d_out, int out_size, void* d_ws, size_t ws_size,
                              hipStream_t stream)
{
    const float* positions = (const float*)d_in[0];   // (N, 3) float32
    const float* table     = (const float*)d_in[1];   // (TOTAL_PARAMS*4,) float32
    (void)n_in; (void)out_size; (void)d_ws; (void)ws_size;

    const int npoints = in_sizes[0] / 3;
    const LevelMeta meta = make_meta();

    const int block  = 256;
    const int blocks = (npoints + block - 1) / block;

    hashgrid_encode_kernel<<<blocks, block, 0, stream>>>(
        positions, (const v4f*)table, (v4f*)d_out, meta, npoints);
}